// EfficientInteractionBilinear_69999376990659
// MI455X (gfx1250) — compile-verified
//
#include <hip/hip_runtime.h>

typedef __attribute__((ext_vector_type(2))) float v2f;
typedef __attribute__((ext_vector_type(8))) float v8f;

#define E_NUM   30000
#define KMAX    16
#define SPH     16
#define EMB_IN  64
#define EMB_C   32
#define EMB_OUT 64
#define KBIG    (EMB_C * EMB_IN)   // 2048
#define EPG     16                 // edges per workgroup
#define NWAVES  8

// ---------------------------------------------------------------------------
// Phase 0a: zero the m_padded workspace (E*K*EMB_IN floats = 122.88 MB)
// grid: 30000 x 256, each thread writes one float4 (7,680,000 total)
__global__ void eib_zero_ws(float4* __restrict__ ws) {
    ws[(size_t)blockIdx.x * 256 + threadIdx.x] = float4{0.f, 0.f, 0.f, 0.f};
}

// Phase 0b: scatter m (T x 64) into m_padded (E x 16 x 64) by (outer, inner)
// grid: 30000 x 256, one float4 per thread (t = gid/16, float4 chunk q = gid%16)
__global__ void eib_scatter_m(const float4* __restrict__ m,
                              const int* __restrict__ outer,
                              const int* __restrict__ inner,
                              float4* __restrict__ mp) {
    size_t gid = (size_t)blockIdx.x * 256 + threadIdx.x;
    int t = (int)(gid >> 4);
    int q = (int)(gid & 15);
    size_t dst = ((size_t)outer[t] * KMAX + inner[t]) * (EMB_IN / 4) + q;
    mp[dst] = m[gid];
}

// ---------------------------------------------------------------------------
// Main fused kernel: per 16-edge group
//   RS = R@S (32x16), rad = RS@Mp (32x64) -> LDS A (16 x 2048)
//   out = A (16x2048) @ W (2048x64)
// All via V_WMMA_F32_16X16X4_F32.
// f32 WMMA operand layouts (wave32):
//   A 16x4 : lane = (h= lane>>4, r = lane&15): a[v] = A[r][2h+v]
//   B 4x16 : b[v] = B[2h+v][lane&15]
//   C/D    : d[v] = D[v + 8h][lane&15]
__global__ __launch_bounds__(256)
void eib_main(const float* __restrict__ radW1,   // (E, 32, 16)
              const float* __restrict__ sph,     // (E, 16, 16)
              const float* __restrict__ mp,      // (E, 16, 64)
              const float* __restrict__ W,       // (2048, 64)
              float* __restrict__ out)           // (E, 64)
{
    __shared__ float ldsA[EPG * KBIG];          // 128 KB: rad rows (A matrix)
    __shared__ float ldsStage[NWAVES * 512];    //  16 KB: per-wave RS / partials

    const int tid  = threadIdx.x;
    const int wave = tid >> 5;
    const int lane = tid & 31;
    const int h    = lane >> 4;
    const int n    = lane & 15;
    const int g    = blockIdx.x;

    // ---------------- Phase A: build A rows (2 edges per wave) --------------
    float* stage = ldsStage + wave * 512;
    for (int eidx = 0; eidx < 2; ++eidx) {
        const int eloc = wave * 2 + eidx;
        const int e    = g * EPG + eloc;
        const float* Re = radW1 + (size_t)e * (EMB_C * SPH);    // 32x16
        const float* Se = sph   + (size_t)e * (SPH * KMAX);     // 16x16
        const float* Me = mp    + (size_t)e * (KMAX * EMB_IN);  // 16x64

        // RS = R @ S  (32x16): two 16-row blocks, K=16 over s
        for (int rb = 0; rb < 2; ++rb) {
            v8f acc = {};
#pragma unroll
            for (int kc = 0; kc < 4; ++kc) {
                const int s0 = kc * 4 + 2 * h;
                v2f a = *(const v2f*)(Re + (rb * 16 + n) * SPH + s0);
                v2f b;
                b.x = Se[s0 * KMAX + n];
                b.y = Se[(s0 + 1) * KMAX + n];
                acc = __builtin_amdgcn_wmma_f32_16x16x4_f32(
                          false, a, false, b, (short)0, acc, false, false);
            }
#pragma unroll
            for (int v = 0; v < 8; ++v)
                stage[(rb * 16 + v + 8 * h) * 16 + n] = acc[v];
        }
        // same-wave RAW on staging: wait LDS stores complete
        asm volatile("s_wait_dscnt 0" ::: "memory");

        // rad = RS @ Mp (32x64) -> ldsA row eloc (flatten c*64 + i)
        for (int rb = 0; rb < 2; ++rb) {
            for (int t4 = 0; t4 < 4; ++t4) {
                v8f acc = {};
#pragma unroll
                for (int kc = 0; kc < 4; ++kc) {
                    const int k0 = kc * 4 + 2 * h;
                    v2f a = *(const v2f*)(stage + (rb * 16 + n) * 16 + k0);
                    v2f b;
                    b.x = Me[k0 * EMB_IN + t4 * 16 + n];
                    b.y = Me[(k0 + 1) * EMB_IN + t4 * 16 + n];
                    acc = __builtin_amdgcn_wmma_f32_16x16x4_f32(
                              false, a, false, b, (short)0, acc, false, false);
                }
#pragma unroll
                for (int v = 0; v < 8; ++v)
                    ldsA[eloc * KBIG + (rb * 16 + v + 8 * h) * EMB_IN
                         + t4 * 16 + n] = acc[v];
            }
        }
    }

    __syncthreads();

    // ------- Phase B: (16 edges x 2048) @ W (2048 x 64), 256 WMMA/wave ------
    const int tile  = wave & 3;   // N tile (cols 16*tile .. +15)
    const int khalf = wave >> 2;  // K half (0: k<1024, 1: k>=1024)
    const float* Wt = W + tile * 16 + n;
    v8f acc = {};
#pragma unroll 4
    for (int kc = 0; kc < 256; ++kc) {
        const int kbase = khalf * 1024 + kc * 4 + 2 * h;
        v2f a = *(const v2f*)(ldsA + n * KBIG + kbase);
        v2f b;
        b.x = Wt[(size_t)kbase * EMB_OUT];
        b.y = Wt[(size_t)(kbase + 1) * EMB_OUT];
        acc = __builtin_amdgcn_wmma_f32_16x16x4_f32(
                  false, a, false, b, (short)0, acc, false, false);
    }
    // partial tile -> own staging slot (no cross-wave hazard)
#pragma unroll
    for (int v = 0; v < 8; ++v)
        stage[(v + 8 * h) * 16 + n] = acc[v];

    __syncthreads();

    // ------------- Reduce 2 K-halves and store 16x64 outputs ----------------
#pragma unroll
    for (int j = 0; j < 4; ++j) {
        const int idx = tid + j * 256;    // 0..1023
        const int row = idx >> 6;         // edge within group
        const int col = idx & 63;         // output column
        const int tt  = col >> 4;
        const int nn  = col & 15;
        const float s = ldsStage[tt * 512 + row * 16 + nn]
                      + ldsStage[(4 + tt) * 512 + row * 16 + nn];
        out[(size_t)(g * EPG + row) * EMB_OUT + col] = s;
    }
}

// ---------------------------------------------------------------------------
extern "C" void kernel_launch(void* const* d_in, const int* in_sizes, int n_in,
                              void* d_out, int out_size, void* d_ws, size_t ws_size,
                              hipStream_t stream) {
    const float* radW1 = (const float*)d_in[0];  // (E, 32, 16)
    const float* sph   = (const float*)d_in[1];  // (E, 16, 16)
    const float* m     = (const float*)d_in[2];  // (T, 64)
    const float* W     = (const float*)d_in[3];  // (2048, 64)
    const int* outer   = (const int*)d_in[4];    // (T,)
    const int* inner   = (const int*)d_in[5];    // (T,)
    float* out = (float*)d_out;                  // (E, 64)
    float* mpad = (float*)d_ws;                  // E*16*64 floats = 122.88 MB

    // E*K*EMB_IN/4 float4 = 7,680,000 -> 30000 blocks x 256 threads exactly
    eib_zero_ws<<<30000, 256, 0, stream>>>((float4*)mpad);
    eib_scatter_m<<<30000, 256, 0, stream>>>((const float4*)m, outer, inner,
                                             (float4*)mpad);
    // 30000 edges / 16 per group = 1875 workgroups
    eib_main<<<1875, 256, 0, stream>>>(radW1, sph, mpad, W, out);
}